// NavierStokesLoss_67774583931055
// MI455X (gfx1250) — compile-verified
//
#include <hip/hip_runtime.h>

// NavierStokesLoss for MI455X (gfx1250), compile-only loop.
// field: (2, 4, 160, 160, 160) fp32; output: scalar fp32 loss.
// CDNA5 data path: Tensor Data Mover (TENSORcnt) stages 34x10 halo tiles of
// all 4 fields into a 4-deep LDS plane ring (one tensor_load_to_lds per field
// per plane, issued by wave 0 only); falls back to per-lane async-to-LDS b32
// (ASYNCcnt) if the TDM builtins are absent. Stencil reads from LDS.
// Deterministic two-pass reduction; v_wmma_f32_16x16x4_f32 wave reduction.

#define N_DIM 160
#define INTER (N_DIM - 2)   // 158 interior points per axis
#define TX 34               // tile + x halo
#define TY 10               // tile + y halo
#define PLANE_F (TX * TY)   // 340 elements per field per plane

typedef __attribute__((ext_vector_type(2))) float v2f;
typedef __attribute__((ext_vector_type(8))) float v8f;
typedef __attribute__((ext_vector_type(4))) unsigned int v4u;
typedef __attribute__((ext_vector_type(4))) int v4i;
typedef __attribute__((ext_vector_type(8))) int v8i;

typedef __attribute__((address_space(1))) int as1_int;   // global
typedef __attribute__((address_space(3))) int as3_int;   // LDS

#if __has_builtin(__builtin_amdgcn_tensor_load_to_lds_d2) || \
    __has_builtin(__builtin_amdgcn_tensor_load_to_lds)
#define HAVE_TDM 1
#else
#define HAVE_TDM 0
#endif

#if HAVE_TDM
// One 2D tile DMA: global (row stride 160 floats) -> contiguous LDS rows.
// padX: tile_dim0=32 with +2 DWORD pad per 32 DWORDs to keep 34-float LDS rows.
__device__ __forceinline__ void tdm_load_tile(const float* g, unsigned ldsByteAddr,
                                              int tileX, int tileY, bool padX)
{
    const unsigned long long ga = (unsigned long long)(size_t)g;

    v4u g0;
    g0.x = 1u;                                       // count=1, user descriptor
    g0.y = ldsByteAddr;                              // lds_addr [63:32]
    g0.z = (unsigned)ga;                             // global_addr[31:0]
    g0.w = (unsigned)((ga >> 32) & 0x01FFFFFFull)    // global_addr[56:32]
           | 0x80000000u;                            // type=2 ("image") [127:126]

    unsigned w0 = 0x20000u;                          // data_size=2 (4 bytes)
    if (padX) w0 |= (1u << 20)                       // pad_enable
                  | (4u << 22)                       // pad_interval: every 32 DWORDs
                  | (1u << 25);                      // pad_amount: 2 DWORDs
    v8i g1;
    g1[0] = (int)w0;                                 // mask=0, no barrier/iterate
    g1[1] = (int)(160u << 16);                       // tensor_dim0=160 (low16 @bit48)
    g1[2] = (int)(160u << 16);                       // dim0 hi=0 | tensor_dim1=160
    g1[3] = (int)((unsigned)tileX << 16);            // dim1 hi=0 | tile_dim0
    g1[4] = (int)(unsigned)tileY;                    // tile_dim1 | tile_dim2=0 (2D)
    g1[5] = 160;                                     // tensor_dim0_stride low32
    g1[6] = (int)(25600u << 16);                     // stride0 hi=0 | dim1_stride=25600
    g1[7] = 0;                                       // dim1_stride hi

#if __has_builtin(__builtin_amdgcn_tensor_load_to_lds_d2)
    __builtin_amdgcn_tensor_load_to_lds_d2(g0, g1, /*cpol=*/0);
#else
    v4i z4 = {0, 0, 0, 0};
#if __clang_major__ >= 23
    v8i z8 = {0, 0, 0, 0, 0, 0, 0, 0};
    __builtin_amdgcn_tensor_load_to_lds(g0, g1, z4, z4, z8, /*cpol=*/0);
#else
    __builtin_amdgcn_tensor_load_to_lds(g0, g1, z4, z4, /*cpol=*/0);
#endif
#endif
}

__device__ __forceinline__ void wait_tensor0()
{
#if __has_builtin(__builtin_amdgcn_s_wait_tensorcnt)
    __builtin_amdgcn_s_wait_tensorcnt(0);
#else
    asm volatile("s_wait_tensorcnt 0" ::: "memory");
#endif
}
#else  // !HAVE_TDM: per-lane async global->LDS staging (proven path)
__device__ __forceinline__ void async_ld_b32(const float* g, float* l)
{
#if __has_builtin(__builtin_amdgcn_global_load_async_to_lds_b32)
    __builtin_amdgcn_global_load_async_to_lds_b32(
        (as1_int*)g, (as3_int*)l, /*offset=*/0, /*cpol=*/0);
#else
    const unsigned lofs =
        (unsigned)(unsigned long long)(__attribute__((address_space(3))) void*)l;
    asm volatile("global_load_async_to_lds_b32 %0, %1, off"
                 : : "v"(lofs), "v"(g) : "memory");
#endif
}
__device__ __forceinline__ void wait_async0()
{
#if __has_builtin(__builtin_amdgcn_s_wait_asynccnt)
    __builtin_amdgcn_s_wait_asynccnt(0);
#else
    asm volatile("s_wait_asynccnt 0" ::: "memory");
#endif
}
#endif

__global__ __launch_bounds__(256) void ns_stencil_kernel(
    const float* __restrict__ field, float* __restrict__ partial, int zChunk)
{
    const float RHO = 1.0f, MU = 0.01f;
    const int tx = threadIdx.x, ty = threadIdx.y;
    const int tid = ty * 32 + tx;
    const int gx0 = 1 + blockIdx.x * 32;       // first interior x of tile
    const int gy0 = 1 + blockIdx.y * 8;        // first interior y of tile
    const int gx = gx0 + tx;                   // this thread's interior x
    const int gy = gy0 + ty;
    const int numChunks = (INTER + zChunk - 1) / zChunk;
    const int b     = blockIdx.z / numChunks;  // batch
    const int chunk = blockIdx.z % numChunks;
    const int z0 = 1 + chunk * zChunk;
    int z1 = z0 + zChunk; if (z1 > INTER + 1) z1 = INTER + 1;   // exclusive

    const int zs = N_DIM * N_DIM;
    const int ys = N_DIM;
    const size_t cs = (size_t)N_DIM * N_DIM * N_DIM;
    const bool valid = (gx <= INTER) && (gy <= INTER);

    // 4-deep plane ring x 4 fields x (10 x 34) halo tile = 21,760 B
    __shared__ float lds[4][4][TY][TX];

#if HAVE_TDM
    // Clamped tile extents (edge blocks): x-edge keeps 34-float row stride via
    // TDM padding; y-edge leaves trailing rows stale (read only by invalid
    // threads). Uniform per block.
    const int tileX = (N_DIM - (gx0 - 1) < TX) ? (N_DIM - (gx0 - 1)) : TX;   // 34 or 32
    const int tileY = (N_DIM - (gy0 - 1) < TY) ? (N_DIM - (gy0 - 1)) : TY;   // 10 or 8
    const bool padX = (tileX != TX);
    // Uniform scalar branch: true only for wave 0 (readfirstlane -> SGPR).
    const bool wave0 = (__builtin_amdgcn_readfirstlane(tid) == 0);

    #define STAGE_PLANE(zz, slot)                                                   \
        do {                                                                        \
            if (wave0) {                                                            \
                const size_t pbase = (size_t)(b * 4) * cs + (size_t)(zz) * zs       \
                                     + (size_t)((gy0 - 1) * ys + (gx0 - 1));        \
                for (int f = 0; f < 4; ++f) {                                       \
                    const float* g = field + pbase + (size_t)f * cs;                \
                    const unsigned la =                                             \
                        (unsigned)(size_t)(as3_int*)&lds[slot][f][0][0];            \
                    tdm_load_tile(g, la, tileX, tileY, padX);                       \
                }                                                                   \
            }                                                                       \
        } while (0)
    #define STAGE_WAIT()  do { if (wave0) wait_tensor0(); } while (0)
#else
    // Per-lane async staging, flat-partitioned over 256 threads.
    #define STAGE_PLANE(zz, slot)                                                   \
        do {                                                                        \
            const size_t fbase = (size_t)(b * 4) * cs + (size_t)(zz) * zs;          \
            for (int i = tid; i < 4 * PLANE_F; i += 256) {                          \
                const int f  = i / PLANE_F;                                         \
                const int r  = i - f * PLANE_F;                                     \
                const int lyy = r / TX;                                             \
                const int lxx = r - lyy * TX;                                       \
                const int gyh = gy0 - 1 + lyy;                                      \
                const int gxh = gx0 - 1 + lxx;                                      \
                if (gyh < N_DIM && gxh < N_DIM) {                                   \
                    const float* g = field + fbase + (size_t)f * cs                 \
                                     + (size_t)(gyh * ys + gxh);                    \
                    async_ld_b32(g, &lds[slot][f][lyy][lxx]);                       \
                }                                                                   \
            }                                                                       \
        } while (0)
    #define STAGE_WAIT()  wait_async0()
#endif

    // Prologue: planes z0-1, z0, z0+1
    STAGE_PLANE(z0 - 1, (z0 - 1) & 3);
    STAGE_PLANE(z0,     (z0)     & 3);
    STAGE_PLANE(z0 + 1, (z0 + 1) & 3);
    STAGE_WAIT();
    __syncthreads();

    float acc = 0.0f;
    const int ly = ty + 1, lx = tx + 1;   // this thread's LDS coords

    for (int z = z0; z < z1; ++z) {
        // overlap: fetch plane z+2 (needed next iteration) into slot (z+2)&3,
        // which held plane z-2 (last read during iteration z-1; the barrier at
        // the end of iteration z-1 makes the overwrite safe).
        if (z + 2 <= z1) STAGE_PLANE(z + 2, (z + 2) & 3);

        if (valid) {
            const int sm = (z - 1) & 3, s0 = z & 3, sp = (z + 1) & 3;

            const float uzc = lds[s0][0][ly][lx];
            const float uyc = lds[s0][1][ly][lx];
            const float uxc = lds[s0][2][ly][lx];

            const float dpz = 0.5f * (lds[sp][3][ly][lx]     - lds[sm][3][ly][lx]);
            const float dpy = 0.5f * (lds[s0][3][ly + 1][lx] - lds[s0][3][ly - 1][lx]);
            const float dpx = 0.5f * (lds[s0][3][ly][lx + 1] - lds[s0][3][ly][lx - 1]);

            // momentum residual: RHO*(u.grad)u + grad(p) - MU*lap(u)
            #define LOSSU(f, c0, dp) ({                                             \
                const float xm = lds[s0][f][ly][lx - 1];                            \
                const float xp = lds[s0][f][ly][lx + 1];                            \
                const float ym = lds[s0][f][ly - 1][lx];                            \
                const float yp = lds[s0][f][ly + 1][lx];                            \
                const float zm = lds[sm][f][ly][lx];                                \
                const float zp = lds[sp][f][ly][lx];                                \
                RHO * (uzc * (0.5f * (zp - zm))                                     \
                     + uyc * (0.5f * (yp - ym))                                     \
                     + uxc * (0.5f * (xp - xm)))                                    \
                 + (dp)                                                             \
                 - MU * ((xp + xm + yp + ym + zp + zm) - 6.0f * (c0)); })

            const float lz = LOSSU(0, uzc, dpz);
            const float lv = LOSSU(1, uyc, dpy);
            const float lh = LOSSU(2, uxc, dpx);
            #undef LOSSU

            acc = fmaf(lz, lz, acc);
            acc = fmaf(lv, lv, acc);
            acc = fmaf(lh, lh, acc);
        }

        STAGE_WAIT();        // staging for plane z+2 complete (issuing wave)
        __syncthreads();     // writes visible to all waves; slot reads done
    }
    #undef STAGE_PLANE
    #undef STAGE_WAIT

    // deterministic block reduction (fixed tree order)
    __shared__ float red[256];
    red[tid] = acc;
    __syncthreads();
    for (int s = 128; s > 0; s >>= 1) {
        if (tid < s) red[tid] += red[tid + s];
        __syncthreads();
    }
    if (tid == 0) {
        const int bid = ((int)blockIdx.z * (int)gridDim.y + (int)blockIdx.y) * (int)gridDim.x
                        + (int)blockIdx.x;
        partial[bid] = red[0];
    }
}

__global__ __launch_bounds__(256) void ns_reduce_kernel(
    const float* __restrict__ partial, int n, float* __restrict__ out, float invN)
{
    __shared__ float red[256];
    const int tid = threadIdx.x;
    float s = 0.0f;
    for (int i = tid; i < n; i += 256) s += partial[i];   // fixed order per lane
    red[tid] = s;
    __syncthreads();

    if (tid < 32) {   // wave 0 only: EXEC all-1s inside this wave (WMMA requirement)
        float v = 0.0f;
        #pragma unroll
        for (int k = 0; k < 8; ++k) v += red[tid + 32 * k];

        // Wave reduction on the matrix pipe:
        // A(16x4): lane L<16 holds A[L][0]=v; lane L>=16 holds A[L-16][2]=v.
        // B = all-ones => D[m][n] = v_m + v_{m+16}.
        // Lane 0's D VGPRs hold D[0..7][0]; lane 16's hold D[8..15][0].
        v2f a;    a.x = v;    a.y = 0.0f;
        v2f ones; ones.x = 1.0f; ones.y = 1.0f;
        v8f cacc = {};
        v8f d = __builtin_amdgcn_wmma_f32_16x16x4_f32(
                    /*neg_a=*/false, a, /*neg_b=*/false, ones,
                    /*c_mod=*/(short)0, cacc, /*reuse_a=*/false, /*reuse_b=*/false);

        float lo = ((d[0] + d[1]) + (d[2] + d[3])) + ((d[4] + d[5]) + (d[6] + d[7]));
        float hi = __shfl(lo, 16, 32);   // lane16's partial (rows 8..15)
        if (tid == 0) out[0] = (lo + hi) * invN;
    }
}

extern "C" void kernel_launch(void* const* d_in, const int* in_sizes, int n_in,
                              void* d_out, int out_size, void* d_ws, size_t ws_size,
                              hipStream_t stream)
{
    (void)in_sizes; (void)n_in; (void)out_size; (void)ws_size;
    const float* field   = (const float*)d_in[0];   // (2,4,160,160,160) fp32
    // d_in[1] ("inp") is unused by the reference.
    float*       out     = (float*)d_out;           // scalar fp32
    float*       partial = (float*)d_ws;            // per-block partial sums

    const int zChunk = 40;
    const int numChunks = (INTER + zChunk - 1) / zChunk;          // 4
    dim3 block(32, 8, 1);                                         // 8 waves (wave32)
    dim3 grid((INTER + 31) / 32, (INTER + 7) / 8, 2 * numChunks); // (5,20,8)=800 blocks

    ns_stencil_kernel<<<grid, block, 0, stream>>>(field, partial, zChunk);

    const int nPartial = (int)(grid.x * grid.y * grid.z);
    const float invN = 1.0f / (2.0f * (float)INTER * (float)INTER * (float)INTER); // 1/7888624
    ns_reduce_kernel<<<1, 256, 0, stream>>>(partial, nPartial, out, invN);
}